// MinibatchDiscrimination_31714038513724
// MI455X (gfx1250) — compile-verified
//
#include <hip/hip_runtime.h>
#include <hip/hip_bf16.h>

// Problem constants (from reference): N=1024, IN_F=512, OUT_F=64, KD=5
#define NROWS 1024
#define INF   512
#define OUTF  64
#define KDIM  5
#define COLS  (OUTF * KDIM)   // 320

typedef __attribute__((ext_vector_type(2))) float v2f;
typedef __attribute__((ext_vector_type(8))) float v8f;

// ---------------------------------------------------------------------------
// Stage 1: Mt[col][n] = sum_k X[n][k] * W[k][col],  W = T flattened (512x320)
// One wave32 computes one 16x16 tile of C via V_WMMA_F32_16X16X4_F32,
// looping K in steps of 4. Result stored TRANSPOSED (col-major over n) so
// stage 2 streams contiguously over j.
// ---------------------------------------------------------------------------
__global__ __launch_bounds__(256)
void mbd_gemm_wmma(const float* __restrict__ X,
                   const float* __restrict__ W,
                   float* __restrict__ Mt)
{
    const int tid       = threadIdx.x;
    const int lane      = tid & 31;
    const int waveInBlk = tid >> 5;
    const int wave      = blockIdx.x * 8 + waveInBlk;   // 1280 waves total
    const int rowTile   = wave / (COLS / 16);           // 0..63
    const int colTile   = wave % (COLS / 16);           // 0..19
    const int row0      = rowTile * 16;
    const int col0      = colTile * 16;

    const int m  = lane & 15;            // matrix row (A) / col (B,C)
    const int kh = (lane >> 4) << 1;     // K offset: lanes 0-15 -> 0, 16-31 -> 2

    v8f acc = {};
    const float* xrow = X + (size_t)(row0 + m) * INF;

    for (int kk = 0; kk < INF; kk += 4) {
        const int kA = kk + kh;
        // A fragment: 16x4 f32.  lane<16: {K=kk+0, kk+1}; lane>=16: {kk+2, kk+3}
        const float2 av = *(const float2*)(xrow + kA);   // 8B aligned (kA even)
        v2f a; a.x = av.x; a.y = av.y;
        // B fragment: 4x16 f32.  lane<16: rows kk+0/kk+1; lane>=16: rows kk+2/kk+3
        v2f b;
        b.x = W[(size_t)kA       * COLS + col0 + m];
        b.y = W[(size_t)(kA + 1) * COLS + col0 + m];
        // D = A*B + C   (8 args: neg_a, A, neg_b, B, c_mod, C, reuse_a, reuse_b)
        acc = __builtin_amdgcn_wmma_f32_16x16x4_f32(
            /*neg_a=*/false, a, /*neg_b=*/false, b,
            /*c_mod=*/(short)0, acc, /*reuse_a=*/false, /*reuse_b=*/false);
    }

    // C/D layout: VGPR v -> row = row0 + v + 8*(lane>=16), col = col0 + m
    const int rbase = row0 + ((lane >> 4) << 3);
    #pragma unroll
    for (int v = 0; v < 8; ++v) {
        Mt[(size_t)(col0 + m) * NROWS + (rbase + v)] = acc[v];
    }
}

// ---------------------------------------------------------------------------
// Stage 2: out[i][o] = sum_j exp(-sum_k |Mt[o*5+k][i] - Mt[o*5+k][j]|)
// Block = (feature o, 256-row i-tile). The 20KB slice Mt[o*5 .. o*5+4][:]
// lives in LDS (320KB/WGP available).
// ---------------------------------------------------------------------------
__global__ __launch_bounds__(256)
void mbd_pairwise(const float* __restrict__ Mt,
                  float* __restrict__ out)
{
    __shared__ float sm[KDIM][NROWS];    // 20 KB

    const int o     = blockIdx.x;        // 0..63
    const int itile = blockIdx.y;        // 0..3
    const int tid   = threadIdx.x;

    #pragma unroll
    for (int k = 0; k < KDIM; ++k)
        for (int j = tid; j < NROWS; j += 256)
            sm[k][j] = Mt[(size_t)(o * KDIM + k) * NROWS + j];
    __syncthreads();

    const int i = itile * 256 + tid;
    float mi[KDIM];
    #pragma unroll
    for (int k = 0; k < KDIM; ++k) mi[k] = sm[k][i];

    float accum = 0.0f;
    for (int j = 0; j < NROWS; ++j) {
        float d = 0.0f;
        #pragma unroll
        for (int k = 0; k < KDIM; ++k)
            d += fabsf(mi[k] - sm[k][j]);
        accum += __expf(-d);             // v_exp_f32 (TRANS pipe, co-issues)
    }
    out[(size_t)i * OUTF + o] = accum;
}

// ---------------------------------------------------------------------------
extern "C" void kernel_launch(void* const* d_in, const int* in_sizes, int n_in,
                              void* d_out, int out_size, void* d_ws, size_t ws_size,
                              hipStream_t stream)
{
    const float* X = (const float*)d_in[0];   // (1024, 512) f32
    const float* T = (const float*)d_in[1];   // (512, 64, 5) f32 == W (512, 320)
    float* out = (float*)d_out;               // (1024, 64) f32
    float* Mt  = (float*)d_ws;                // needs 320*1024*4 = 1.25 MB scratch

    // Stage 1: 1280 wave-tiles / 8 waves per block = 160 blocks
    mbd_gemm_wmma<<<160, 256, 0, stream>>>(X, T, Mt);
    // Stage 2: 64 features x 4 row-tiles
    mbd_pairwise<<<dim3(OUTF, NROWS / 256), 256, 0, stream>>>(Mt, out);
}